// PAWNN_62912680952394
// MI455X (gfx1250) — compile-verified
//
#include <hip/hip_runtime.h>
#include <hip/hip_bf16.h>

// ---------------------------------------------------------------------------
// PAWNN block-sparse MLP for MI455X (gfx1250), f32 WMMA path.
//   HIDDEN = 96 buses * 64 = 6144, BATCH = 4096
//   L0:    per bus  [4096 x 30] @ [30 x 64]   (21 own + 9 global features)
//   L1-3:  per bus  [4096 x 192] @ [192 x 64] (ring band: b-1, b, b+1)
//   Lout:  per bus  [4096 x 64] @ [64 x 21]
// ~32 GFLOP effective -> HBM/L2 bound, so f32 WMMA (16x16x4) keeps full
// reference precision at zero throughput cost. Global->LDS staging uses the
// CDNA5 async-to-LDS path (ASYNCcnt) when the toolchain exposes it.
// ---------------------------------------------------------------------------

typedef __attribute__((ext_vector_type(2))) float v2f;
typedef __attribute__((ext_vector_type(8))) float v8f;
typedef __attribute__((ext_vector_type(4))) float f4;
typedef int v4i __attribute__((vector_size(4 * sizeof(int))));

#define NB    96
#define HPB   64
#define HID   6144
#define NIN   2048
#define NOUT  2016
#define NBATCH 4096
#define FPB   21
#define OPB   21
#define NEG_SLOPE 0.01f

#if defined(__has_builtin)
#  if __has_builtin(__builtin_amdgcn_global_load_async_to_lds_b128)
#    define HAVE_ASYNC_B128 1
#  endif
#  if __has_builtin(__builtin_amdgcn_s_wait_asynccnt)
#    define HAVE_WAIT_ASYNC 1
#  endif
#endif
#ifndef HAVE_ASYNC_B128
#  define HAVE_ASYNC_B128 0
#endif
#ifndef HAVE_WAIT_ASYNC
#  define HAVE_WAIT_ASYNC 0
#endif

__device__ __forceinline__ float leaky(float v) {
    return v >= 0.0f ? v : NEG_SLOPE * v;
}

// 16-byte global -> LDS copy. Async (no VGPR round-trip) when available.
__device__ __forceinline__ void copy16_g2l(const float* gsrc, float* ldst) {
#if HAVE_ASYNC_B128
    __builtin_amdgcn_global_load_async_to_lds_b128(
        (__attribute__((address_space(1))) v4i*)gsrc,
        (__attribute__((address_space(3))) v4i*)ldst,
        0, 0);
#else
    *(f4*)ldst = *(const f4*)gsrc;
#endif
}

// Wait for all outstanding async global->LDS transfers from this wave.
__device__ __forceinline__ void async_join() {
#if HAVE_ASYNC_B128
#  if HAVE_WAIT_ASYNC
    __builtin_amdgcn_s_wait_asynccnt(0);
#  else
    asm volatile("s_wait_asynccnt 0x0" ::: "memory");
#  endif
#endif
}

// ---------------------------------------------------------------------------
// Hidden layer: hout[:, bus*64 .. +64] = leaky(band(hin) @ W_band^T + bias)
// grid = (96, NBATCH/64), block = 512 (16 waves; wave w -> Mtile w&3, Ntile w>>2)
// dynamic LDS: sA[64][200] act band, sB[64][200] weight band
// ---------------------------------------------------------------------------
__global__ void pawnn_hidden_kernel(const float* __restrict__ hin,
                                    const float* __restrict__ W,
                                    const float* __restrict__ bias,
                                    float* __restrict__ hout)
{
    extern __shared__ float smem[];
    const int STR = 200;                 // 192 + 8 pad (800B rows, 16B aligned)
    float* sA = smem;                    // [64][STR]
    float* sB = smem + 64 * STR;         // [64][STR]

    const int bus   = blockIdx.x;
    const int mbase = blockIdx.y * 64;
    const int tid   = threadIdx.x;       // 0..511

    // Cooperative load of the 64x192 activation band and weight band as float4.
    // K ordering: [block bus-1 | block bus | block bus+1], 64 cols each.
    #pragma unroll
    for (int it = 0; it < 6; ++it) {
        int i   = tid + it * 512;        // 0 .. 3071 (64 rows * 48 float4)
        int row = i / 48;
        int c4  = i % 48;
        int seg = c4 / 16;               // neighbor 0,1,2
        int off = (c4 % 16) * 4;         // col within the 64-wide block
        int gblk = (bus + 95 + seg) % 96;
        int gcol = gblk * 64 + off;
        int j    = c4 * 4;               // LDS column (seg*64 + off)
        copy16_g2l(hin + (size_t)(mbase + row) * HID + gcol, sA + row * STR + j);
        copy16_g2l(W   + (size_t)(bus * 64 + row) * HID + gcol, sB + row * STR + j);
    }
    async_join();
    __syncthreads();

    const int wave = tid >> 5;           // 0..15
    const int lane = tid & 31;
    const int half = lane >> 4;          // 0: K 0,1   1: K 2,3
    const int l    = lane & 15;
    const int mt   = wave & 3;           // M tile 0..3
    const int nt   = wave >> 2;          // N tile 0..3

    const float bval = bias[bus * 64 + nt * 16 + l];
    v8f acc;
    #pragma unroll
    for (int i = 0; i < 8; ++i) acc[i] = bval;

    const float* arow = sA + (mt * 16 + l) * STR + 2 * half;
    const float* brow = sB + (nt * 16 + l) * STR + 2 * half;

    #pragma unroll
    for (int k = 0; k < 192; k += 4) {
        v2f a, b;
        a[0] = arow[k]; a[1] = arow[k + 1];     // ds_load b64/2addr
        b[0] = brow[k]; b[1] = brow[k + 1];
        acc = __builtin_amdgcn_wmma_f32_16x16x4_f32(
            false, a, false, b, (short)0, acc, false, false);
    }

    // D layout: acc[i] -> row (half*8 + i), col l within the 16x16 tile.
    float* orow = hout + (size_t)(mbase + mt * 16 + half * 8) * HID
                       + bus * 64 + nt * 16 + l;
    #pragma unroll
    for (int i = 0; i < 8; ++i) {
        orow[(size_t)i * HID] = leaky(acc[i]);
    }
}

// ---------------------------------------------------------------------------
// Layer 0: per bus, K = 30 (21 own + 9 global) padded to 32.
// grid = (96, NBATCH/64), block = 512; LDS sA[64][40], sB[64][40]
// Irregular column gather (21 own + 9 global + zero pad) -> VGPR staging.
// ---------------------------------------------------------------------------
__global__ void pawnn_layer0_kernel(const float* __restrict__ x,
                                    const float* __restrict__ W0,
                                    const float* __restrict__ b0,
                                    float* __restrict__ hout)
{
    extern __shared__ float smem[];
    const int STR = 40;
    float* sA = smem;                    // [64][40]
    float* sB = smem + 64 * STR;         // [64][40]

    const int bus   = blockIdx.x;
    const int mbase = blockIdx.y * 64;
    const int tid   = threadIdx.x;

    // 64 rows x 32 K-cols each matrix, scalar gather (cols 30,31 zero-padded).
    #pragma unroll
    for (int it = 0; it < 4; ++it) {
        int i   = tid + it * 512;        // 0..2047
        int row = i >> 5;
        int j   = i & 31;
        float xv = 0.0f, wv = 0.0f;
        if (j < FPB) {
            int gc = bus * FPB + j;
            xv = x [(size_t)(mbase + row)    * NIN + gc];
            wv = W0[(size_t)(bus * 64 + row) * NIN + gc];
        } else if (j < FPB + 9) {
            int gc = (NIN - 9) + (j - FPB);
            xv = x [(size_t)(mbase + row)    * NIN + gc];
            wv = W0[(size_t)(bus * 64 + row) * NIN + gc];
        }
        sA[row * STR + j] = xv;
        sB[row * STR + j] = wv;
    }
    __syncthreads();

    const int wave = tid >> 5;
    const int lane = tid & 31;
    const int half = lane >> 4;
    const int l    = lane & 15;
    const int mt   = wave & 3;
    const int nt   = wave >> 2;

    const float bval = b0[bus * 64 + nt * 16 + l];
    v8f acc;
    #pragma unroll
    for (int i = 0; i < 8; ++i) acc[i] = bval;

    const float* arow = sA + (mt * 16 + l) * STR + 2 * half;
    const float* brow = sB + (nt * 16 + l) * STR + 2 * half;

    #pragma unroll
    for (int k = 0; k < 32; k += 4) {
        v2f a, b;
        a[0] = arow[k]; a[1] = arow[k + 1];
        b[0] = brow[k]; b[1] = brow[k + 1];
        acc = __builtin_amdgcn_wmma_f32_16x16x4_f32(
            false, a, false, b, (short)0, acc, false, false);
    }

    float* orow = hout + (size_t)(mbase + mt * 16 + half * 8) * HID
                       + bus * 64 + nt * 16 + l;
    #pragma unroll
    for (int i = 0; i < 8; ++i) {
        orow[(size_t)i * HID] = leaky(acc[i]);
    }
}

// ---------------------------------------------------------------------------
// Output layer: out[:, bus*21 .. +21] = hin[:, bus*64 .. +64] @ Wo_b^T + bo
// N padded 21 -> 32 (2 N-tiles, masked store). grid = (96, NBATCH/64),
// block = 256 (8 waves: 4 M-tiles x 2 N-tiles). LDS sA[64][72], sB[32][72].
// ---------------------------------------------------------------------------
__global__ void pawnn_out_kernel(const float* __restrict__ hin,
                                 const float* __restrict__ Wo,
                                 const float* __restrict__ bo,
                                 float* __restrict__ out)
{
    extern __shared__ float smem[];
    const int STR = 72;
    float* sA = smem;                    // [64][72]
    float* sB = smem + 64 * STR;         // [32][72]

    const int bus   = blockIdx.x;
    const int mbase = blockIdx.y * 64;
    const int tid   = threadIdx.x;       // 0..255

    // sA: 64x64 activations of this bus, float4 (async when available).
    #pragma unroll
    for (int it = 0; it < 4; ++it) {
        int i   = tid + it * 256;        // 0..1023 (64 rows * 16 float4)
        int row = i >> 4;
        int c4  = (i & 15) * 4;
        copy16_g2l(hin + (size_t)(mbase + row) * HID + bus * 64 + c4,
                   sA + row * STR + c4);
    }
    // sB: 32x64 weights (rows 21..31 zero).
    #pragma unroll
    for (int it = 0; it < 2; ++it) {
        int i   = tid + it * 256;        // 0..511 (32 rows * 16 float4)
        int row = i >> 4;
        int c4  = (i & 15) * 4;
        if (row < OPB) {
            copy16_g2l(Wo + (size_t)(bus * OPB + row) * HID + bus * 64 + c4,
                       sB + row * STR + c4);
        } else {
            f4 z = {0.0f, 0.0f, 0.0f, 0.0f};
            *(f4*)(sB + row * STR + c4) = z;   // ds_store, DScnt path
        }
    }
    async_join();
    __syncthreads();

    const int wave = tid >> 5;           // 0..7
    const int lane = tid & 31;
    const int half = lane >> 4;
    const int l    = lane & 15;
    const int mt   = wave & 3;           // 4 M tiles
    const int nt   = wave >> 2;          // 2 N tiles

    const int ncol = nt * 16 + l;        // 0..31
    const float bval = (ncol < OPB) ? bo[bus * OPB + ncol] : 0.0f;
    v8f acc;
    #pragma unroll
    for (int i = 0; i < 8; ++i) acc[i] = bval;

    const float* arow = sA + (mt * 16 + l) * STR + 2 * half;
    const float* brow = sB + ncol * STR + 2 * half;

    #pragma unroll
    for (int k = 0; k < 64; k += 4) {
        v2f a, b;
        a[0] = arow[k]; a[1] = arow[k + 1];
        b[0] = brow[k]; b[1] = brow[k + 1];
        acc = __builtin_amdgcn_wmma_f32_16x16x4_f32(
            false, a, false, b, (short)0, acc, false, false);
    }

    if (ncol < OPB) {
        float* orow = out + (size_t)(mbase + mt * 16 + half * 8) * NOUT
                          + bus * OPB + ncol;
        #pragma unroll
        for (int i = 0; i < 8; ++i) {
            orow[(size_t)i * NOUT] = acc[i];   // no activation on final layer
        }
    }
}

// ---------------------------------------------------------------------------
// Launch: x -> buf0 -> buf1 -> buf0 -> buf1 -> out  (activations ping-pong
// through d_ws; 2 * 4096*6144 floats = 201 MB of workspace).
// ---------------------------------------------------------------------------
extern "C" void kernel_launch(void* const* d_in, const int* in_sizes, int n_in,
                              void* d_out, int out_size, void* d_ws, size_t ws_size,
                              hipStream_t stream) {
    const float* x  = (const float*)d_in[0];
    const float* W0 = (const float*)d_in[1];
    const float* b0 = (const float*)d_in[2];
    const float* W1 = (const float*)d_in[3];
    const float* b1 = (const float*)d_in[4];
    const float* W2 = (const float*)d_in[5];
    const float* b2 = (const float*)d_in[6];
    const float* W3 = (const float*)d_in[7];
    const float* b3 = (const float*)d_in[8];
    const float* Wo = (const float*)d_in[9];
    const float* bo = (const float*)d_in[10];
    // d_in[11..15] are the masks -- structure is hardcoded, not needed.

    float* out  = (float*)d_out;
    float* buf0 = (float*)d_ws;
    float* buf1 = buf0 + (size_t)NBATCH * HID;

    dim3 grid(NB, NBATCH / 64);
    const size_t lds0   = (size_t)2 * 64 * 40  * sizeof(float);   //  20 KB
    const size_t ldsH   = (size_t)2 * 64 * 200 * sizeof(float);   // 100 KB
    const size_t ldsOut = (size_t)(64 + 32) * 72 * sizeof(float); //  27 KB

    pawnn_layer0_kernel<<<grid, 512, lds0, stream>>>(x, W0, b0, buf0);
    pawnn_hidden_kernel<<<grid, 512, ldsH, stream>>>(buf0, W1, b1, buf1);
    pawnn_hidden_kernel<<<grid, 512, ldsH, stream>>>(buf1, W2, b2, buf0);
    pawnn_hidden_kernel<<<grid, 512, ldsH, stream>>>(buf0, W3, b3, buf1);
    pawnn_out_kernel   <<<grid, 256, ldsOut, stream>>>(buf1, Wo, bo, out);
}